// yoloLoss_13520557047897
// MI455X (gfx1250) — compile-verified
//
#include <hip/hip_runtime.h>

// ---------------------------------------------------------------------------
// YOLO-v1 loss on MI455X (gfx1250).
// Memory-bound: ~96 MB streamed, ~0.12 GFLOP -> HBM roofline ~4.1 us @23.3TB/s.
// Strategy: cooperative async global->LDS staging (global_load_async_to_lds_b128,
// GVS addressing: uniform SGPR base + 32-bit VGPR offset), double buffered;
// per-cell math out of LDS (conflict-free stride-30 across 64 banks);
// deterministic two-pass reduction (no float atomics -> bitwise reproducible).
// ---------------------------------------------------------------------------

#define SGRID 7
#define DCH   30
#define BATCH 8192
#define TILE  128                         // cells per tile (== blockDim.x)
#define BLOCK 128
#define NCELL (BATCH * SGRID * SGRID)     // 401408
#define NT    (NCELL / TILE)              // 3136 tiles, exact
#define NB    1568                        // blocks; 3136/1568 = 2 tiles/block, exact
#define F4_PER_ARR     (TILE * DCH / 4)   // 960 float4 per array per tile
#define FLOATS_PER_ARR (TILE * DCH)       // 3840
#define ARR_BYTES      (FLOATS_PER_ARR * 4)   // 15360 bytes per array per tile

// Issue async b128 global->LDS copies for one tile (pred then target).
// GVS mode: saddr (uniform SGPR pair) + 32-bit byte offset in VGPR.
// LDS layout (single __shared__ object, offset 0 in group segment):
//   sh[buf][0][3840] = pred tile, sh[buf][1][3840] = target tile
__device__ __forceinline__ void async_tile_load(const float* __restrict__ pred,
                                                const float* __restrict__ targ,
                                                int tile, int buf, int tid) {
  const unsigned ldsBuf = (unsigned)buf * (2u * ARR_BYTES);
  const unsigned gTileBytes = (unsigned)tile * ARR_BYTES;   // < 2^31, fits u32
  #pragma unroll
  for (int arr = 0; arr < 2; ++arr) {                       // uniform: SGPR base
    const float* src = (arr == 0) ? pred : targ;
    const unsigned ldsArr = ldsBuf + (unsigned)arr * ARR_BYTES;
    #pragma unroll 1
    for (int k = tid; k < F4_PER_ARR; k += BLOCK) {         // 7.5 iters/thread
      const unsigned lds  = ldsArr + (unsigned)k * 16u;
      const unsigned voff = gTileBytes + (unsigned)k * 16u;
      asm volatile("global_load_async_to_lds_b128 %0, %1, %2"
                   :: "v"(lds), "v"(voff), "s"(src) : "memory");
    }
  }
}

__device__ __forceinline__ float iou_pair(float tx, float ty, float tw, float th,
                                          float px, float py, float pw, float ph,
                                          float fi, float fj) {
  const float STEP = 1.0f / 7.0f;
  // _convert(t)
  float x1 = (tx + fj) * STEP;
  float y1 = (ty + fi) * STEP;
  float a1 = fmaxf(x1 - tw * 0.5f, 0.0f);
  float b1 = fmaxf(y1 - th * 0.5f, 0.0f);
  // _convert(p)
  float x2 = (px + fj) * STEP;
  float y2 = (py + fi) * STEP;
  float a2 = fmaxf(x2 - pw * 0.5f, 0.0f);
  float b2 = fmaxf(y2 - ph * 0.5f, 0.0f);
  float iw = tw + pw - (fmaxf(a1 + tw, a2 + pw) - fminf(a1, a2));
  float ih = th + ph - (fmaxf(b1 + th, b2 + ph) - fminf(b1, b2));
  bool valid = (iw > 0.0f) && (ih > 0.0f);
  float inter = valid ? iw * ih : 0.0f;
  float uni = tw * th + pw * ph - inter;
  return valid ? (inter / uni) : 0.0f;
}

__global__ __launch_bounds__(BLOCK) void yolo_main(const float* __restrict__ pred,
                                                   const float* __restrict__ targ,
                                                   float* __restrict__ partial) {
  __shared__ float sh[2][2][FLOATS_PER_ARR];   // 60 KB, only shared object
  const int tid = threadIdx.x;

  float sObj = 0.f, sNoobj = 0.f, sXYWH = 0.f, sCls = 0.f;

  int t = (int)blockIdx.x;
  int buf = 0;
  if (t < NT) async_tile_load(pred, targ, t, 0, tid);

  for (; t < NT; t += NB) {
    asm volatile("s_wait_asynccnt 0" ::: "memory");  // my wave's copies landed
    __syncthreads();                                  // all waves' copies landed
    const int tn = t + NB;
    if (tn < NT) async_tile_load(pred, targ, tn, buf ^ 1, tid);  // prefetch next

    // ---- compute one cell from LDS (stride-30 -> bank-conflict-free) ----
    const float* pp = &sh[buf][0][tid * DCH];
    const float* tp = &sh[buf][1][tid * DCH];
    const int c  = t * TILE + tid;
    const int cj = c % SGRID;
    const int ci = (c / SGRID) % SGRID;
    const float fj = (float)cj, fi = (float)ci;

    const float t0 = tp[0], t1 = tp[1], t2 = tp[2], t3 = tp[3], t4 = tp[4];
    const float p0 = pp[0], p1 = pp[1], p2 = pp[2], p3 = pp[3];
    const float p5 = pp[5], p6 = pp[6], p7 = pp[7], p8 = pp[8];

    const float iou1 = iou_pair(t0, t1, t2, t3, p0, p1, p2, p3, fi, fj);
    const float iou2 = iou_pair(t0, t1, t2, t3, p5, p6, p7, p8, fi, fj);

    const bool obj = (t4 > 0.0f);
    const bool noobj = (t4 == 0.0f);
    const bool choose1 = (iou1 > iou2);
    const bool m1 = obj && choose1 && (iou1 != 0.0f);
    const bool m2 = obj && !choose1 && (iou2 != 0.0f);

    if (m1) {
      float d = iou1 - pp[4];
      sObj += d * d;
      float dx = t0 - p0, dy = t1 - p1;
      float dw = sqrtf(t2) - sqrtf(p2), dh = sqrtf(t3) - sqrtf(p3);
      sXYWH += dx * dx + dy * dy + dw * dw + dh * dh;
    } else if (m2) {
      float d = iou2 - pp[9];
      sObj += d * d;
      float dx = tp[5] - p5, dy = tp[6] - p6;
      float dw = sqrtf(tp[7]) - sqrtf(p7), dh = sqrtf(tp[8]) - sqrtf(p8);
      sXYWH += dx * dx + dy * dy + dw * dw + dh * dh;
    }
    if (noobj) {
      #pragma unroll
      for (int q = 0; q < 6; ++q) {          // conf idx = 4,9,14,19,24,29
        float d = tp[4 + 5 * q] - pp[4 + 5 * q];
        sNoobj += d * d;
      }
    }
    if (obj) {
      #pragma unroll
      for (int k = 10; k < 30; ++k) {        // 20 classes
        float d = pp[k] - tp[k];
        sCls += d * d;
      }
    }

    __syncthreads();    // buf fully consumed before it is re-filled
    buf ^= 1;
  }

  // ---- deterministic block reduction: wave shuffles, then LDS ----
  #pragma unroll
  for (int o = 16; o > 0; o >>= 1) {
    sObj   += __shfl_down(sObj, o, 32);
    sNoobj += __shfl_down(sNoobj, o, 32);
    sXYWH  += __shfl_down(sXYWH, o, 32);
    sCls   += __shfl_down(sCls, o, 32);
  }
  __syncthreads();                            // sh no longer needed; reuse it
  float* red = &sh[0][0][0];
  const int wave = tid >> 5, lane = tid & 31;
  if (lane == 0) {
    red[wave * 4 + 0] = sObj;
    red[wave * 4 + 1] = sNoobj;
    red[wave * 4 + 2] = sXYWH;
    red[wave * 4 + 3] = sCls;
  }
  __syncthreads();
  if (tid == 0) {
    float a = 0.f, b = 0.f, c2 = 0.f, d2 = 0.f;
    #pragma unroll
    for (int w = 0; w < BLOCK / 32; ++w) {
      a  += red[w * 4 + 0];
      b  += red[w * 4 + 1];
      c2 += red[w * 4 + 2];
      d2 += red[w * 4 + 3];
    }
    partial[blockIdx.x * 4 + 0] = a;
    partial[blockIdx.x * 4 + 1] = b;
    partial[blockIdx.x * 4 + 2] = c2;
    partial[blockIdx.x * 4 + 3] = d2;
  }
}

__global__ __launch_bounds__(256) void yolo_finalize(const float* __restrict__ partial,
                                                     float* __restrict__ out) {
  __shared__ float red[8][4];
  const int tid = threadIdx.x;
  float a = 0.f, b = 0.f, c = 0.f, d = 0.f;
  for (int i = tid; i < NB; i += 256) {
    a += partial[i * 4 + 0];
    b += partial[i * 4 + 1];
    c += partial[i * 4 + 2];
    d += partial[i * 4 + 3];
  }
  #pragma unroll
  for (int o = 16; o > 0; o >>= 1) {
    a += __shfl_down(a, o, 32);
    b += __shfl_down(b, o, 32);
    c += __shfl_down(c, o, 32);
    d += __shfl_down(d, o, 32);
  }
  const int wave = tid >> 5, lane = tid & 31;
  if (lane == 0) { red[wave][0] = a; red[wave][1] = b; red[wave][2] = c; red[wave][3] = d; }
  __syncthreads();
  if (tid == 0) {
    float A = 0.f, B = 0.f, C = 0.f, D = 0.f;
    #pragma unroll
    for (int w = 0; w < 8; ++w) {
      A += red[w][0]; B += red[w][1]; C += red[w][2]; D += red[w][3];
    }
    const float invB = 1.0f / (float)BATCH;
    out[0] = (A + 0.5f * B) * invB;   // conf_loss = (obj + 0.5*noobj)/B
    out[1] = 5.0f * C * invB;         // reg_loss  = 5*(xy+wh)/B
    out[2] = D * invB;                // cls_loss
  }
}

extern "C" void kernel_launch(void* const* d_in, const int* in_sizes, int n_in,
                              void* d_out, int out_size, void* d_ws, size_t ws_size,
                              hipStream_t stream) {
  (void)in_sizes; (void)n_in; (void)out_size; (void)ws_size;
  const float* pred = (const float*)d_in[0];
  const float* targ = (const float*)d_in[1];
  float* partial = (float*)d_ws;                 // NB*4 floats = 25 KB scratch
  yolo_main<<<NB, BLOCK, 0, stream>>>(pred, targ, partial);
  yolo_finalize<<<1, 256, 0, stream>>>(partial, (float*)d_out);
}